// ICNN_38998303048271
// MI455X (gfx1250) — compile-verified
//
#include <hip/hip_runtime.h>

// ICNN gradient (forward + analytic backward), fully fused, fp32 WMMA path.
// B=8192, D=128, H=512, L=10. One workgroup = 16 batch rows, 8 waves.

#define B_TOT   8192
#define D_DIM   128
#define H_DIM   512
#define L_LAY   10
#define NEG     0.2f
#define MU      0.5f
#define ROWS    16
#define NTHREADS 256

typedef float v2f __attribute__((ext_vector_type(2)));
typedef float v8f __attribute__((ext_vector_type(8)));

__device__ __forceinline__ v8f wmma_f32_k4(v2f a, v2f b, v8f c) {
  // D(16x16,f32) = A(16x4,f32) * B(4x16,f32) + C ; full fp32 accumulate
  return __builtin_amdgcn_wmma_f32_16x16x4_f32(false, a, false, b,
                                               (short)0, c, false, false);
}

// LDS layout (bytes)
#define XS_OFF     0          //  16*128 f = 8 KB
#define BUFA_OFF   8192       //  16*512 f = 32 KB  (z / g ping)
#define BUFB_OFF   40960      //  16*512 f = 32 KB  (z / g pong)
#define EBUF_OFF   73728      //  16*512 f = 32 KB  (e = 2*gp*u)
#define GRADX_OFF  106496     //  16*128 f = 8 KB
#define SBUF_OFF   114688     //  16 f
#define MASK_OFF   114752     //  10*16*512 B = 80 KB
#define SMEM_TOTAL 196672

__global__ __launch_bounds__(NTHREADS, 1)
void icnn_grad_kernel(const float* __restrict__ x,
                      const float* __restrict__ Wq,
                      const float* __restrict__ Wl,
                      const float* __restrict__ bl,
                      const float* __restrict__ Wz,
                      const float* __restrict__ wz_out,
                      const float* __restrict__ wq_out,
                      const float* __restrict__ wl_out,
                      float* __restrict__ out) {
  extern __shared__ char smem[];
  float* xs    = (float*)(smem + XS_OFF);
  float* bufA  = (float*)(smem + BUFA_OFF);
  float* bufB  = (float*)(smem + BUFB_OFF);
  float* eBuf  = (float*)(smem + EBUF_OFF);
  float* gradx = (float*)(smem + GRADX_OFF);
  float* sbuf  = (float*)(smem + SBUF_OFF);
  unsigned char* masks = (unsigned char*)(smem + MASK_OFF);

  const int tid  = threadIdx.x;
  const int lane = tid & 31;
  const int wave = tid >> 5;
  const int idx  = lane & 15;   // M (A) / N (B,C) index
  const int half = lane >> 4;   // K-pair selector / M-half selector
  const int row0 = blockIdx.x * ROWS;

  // ---- stage x tile ----
  for (int i = tid; i < ROWS * D_DIM; i += NTHREADS)
    xs[i] = x[row0 * D_DIM + i];
  __syncthreads();

  float* zc = bufA;  // z current
  float* zn = bufB;  // z next

  // ================= forward =================
  for (int l = 0; l < L_LAY; ++l) {
    const float* wq = Wq + (size_t)l * H_DIM * D_DIM;
    const float* wl = Wl + (size_t)l * H_DIM * D_DIM;
    const float* wz = Wz + (size_t)(l - 1) * H_DIM * H_DIM;  // used only l>0
    for (int t = 0; t < 4; ++t) {
      const int col = (wave * 4 + t) * 16 + idx;     // output column in H
      v8f cz = {};
      if (l > 0) {
        // pre += z @ Wz^T : B[k][n] = Wz[col][k]  (contiguous float2)
        const float* zrow = zc + idx * H_DIM + 2 * half;
        const float* wrow = wz + (size_t)col * H_DIM + 2 * half;
        #pragma unroll 4
        for (int k = 0; k < H_DIM; k += 4) {
          v2f a = { zrow[k], zrow[k + 1] };
          v2f b = { wrow[k], wrow[k + 1] };
          cz = wmma_f32_k4(a, b, cz);
        }
      }
      v8f cu = {}, cv = {};
      {
        // u = x @ Wq^T ; v = x @ Wl^T  (K = D = 128)
        const float* xrow = xs + idx * D_DIM + 2 * half;
        const float* qrow = wq + (size_t)col * D_DIM + 2 * half;
        const float* lrow = wl + (size_t)col * D_DIM + 2 * half;
        #pragma unroll 4
        for (int k = 0; k < D_DIM; k += 4) {
          v2f a  = { xrow[k], xrow[k + 1] };
          v2f bq = { qrow[k], qrow[k + 1] };
          v2f bv = { lrow[k], lrow[k + 1] };
          cu = wmma_f32_k4(a, bq, cu);
          cv = wmma_f32_k4(a, bv, cv);
        }
      }
      const float blv = bl[l * H_DIM + col];
      #pragma unroll
      for (int r = 0; r < 8; ++r) {
        const int m = r + 8 * half;                  // batch row in tile
        const float pre = cz[r] + cu[r] * cu[r] + cv[r] + blv;
        const bool pos = pre > 0.0f;
        zn[m * H_DIM + col] = pos ? pre : NEG * pre;
        masks[(l * ROWS + m) * H_DIM + col] = pos ? 1 : 0;
      }
    }
    __syncthreads();
    float* tmp = zc; zc = zn; zn = tmp;
  }

  // ================= backward init =================
  float* gc = bufA;  // g current (reuse z buffers)
  float* gn = bufB;
  for (int i = tid; i < ROWS * H_DIM; i += NTHREADS)
    gc[i] = wz_out[i & (H_DIM - 1)];                 // dL/dz_L = wz_out
  for (int i = tid; i < ROWS * D_DIM; i += NTHREADS)
    gradx[i] = 0.0f;
  __syncthreads();

  // ================= backward =================
  for (int l = L_LAY - 1; l >= 0; --l) {
    const float* wq = Wq + (size_t)l * H_DIM * D_DIM;
    const float* wl = Wl + (size_t)l * H_DIM * D_DIM;

    // P1: recompute u = x@Wq^T, then gp = g*mask (in place), e = 2*gp*u
    for (int t = 0; t < 4; ++t) {
      const int col = (wave * 4 + t) * 16 + idx;
      v8f cu = {};
      const float* xrow = xs + idx * D_DIM + 2 * half;
      const float* qrow = wq + (size_t)col * D_DIM + 2 * half;
      #pragma unroll 4
      for (int k = 0; k < D_DIM; k += 4) {
        v2f a  = { xrow[k], xrow[k + 1] };
        v2f bq = { qrow[k], qrow[k + 1] };
        cu = wmma_f32_k4(a, bq, cu);
      }
      #pragma unroll
      for (int r = 0; r < 8; ++r) {
        const int m  = r + 8 * half;
        const float g  = gc[m * H_DIM + col];
        const float sc = masks[(l * ROWS + m) * H_DIM + col] ? 1.0f : NEG;
        const float gp = g * sc;
        gc[m * H_DIM + col]   = gp;
        eBuf[m * H_DIM + col] = 2.0f * gp * cu[r];
      }
    }
    __syncthreads();

    // P2a: grad_x += gp @ Wl[l] + e @ Wq[l]   (K = H, one 16-col tile/wave)
    {
      const int col = wave * 16 + idx;               // column in D
      v8f cg = {};
      #pragma unroll
      for (int r = 0; r < 8; ++r)
        cg[r] = gradx[(r + 8 * half) * D_DIM + col];
      const float* grow = gc   + idx * H_DIM + 2 * half;
      const float* erow = eBuf + idx * H_DIM + 2 * half;
      #pragma unroll 2
      for (int k = 0; k < H_DIM; k += 4) {
        const int k0 = k + 2 * half;
        v2f a1 = { grow[k], grow[k + 1] };
        v2f b1 = { wl[(size_t)k0 * D_DIM + col], wl[(size_t)(k0 + 1) * D_DIM + col] };
        cg = wmma_f32_k4(a1, b1, cg);
        v2f a2 = { erow[k], erow[k + 1] };
        v2f b2 = { wq[(size_t)k0 * D_DIM + col], wq[(size_t)(k0 + 1) * D_DIM + col] };
        cg = wmma_f32_k4(a2, b2, cg);
      }
      #pragma unroll
      for (int r = 0; r < 8; ++r)
        gradx[(r + 8 * half) * D_DIM + col] = cg[r];
    }

    // P2b: g_{l-1} = gp @ Wz[l-1]   (B[k][n] = Wz[k][n], strided)
    if (l > 0) {
      const float* wz = Wz + (size_t)(l - 1) * H_DIM * H_DIM;
      for (int t = 0; t < 4; ++t) {
        const int col = (wave * 4 + t) * 16 + idx;
        v8f cgn = {};
        const float* grow = gc + idx * H_DIM + 2 * half;
        #pragma unroll 2
        for (int k = 0; k < H_DIM; k += 4) {
          const int k0 = k + 2 * half;
          v2f a = { grow[k], grow[k + 1] };
          v2f b = { wz[(size_t)k0 * H_DIM + col], wz[(size_t)(k0 + 1) * H_DIM + col] };
          cgn = wmma_f32_k4(a, b, cgn);
        }
        #pragma unroll
        for (int r = 0; r < 8; ++r)
          gn[(r + 8 * half) * H_DIM + col] = cgn[r];
      }
    }
    __syncthreads();
    float* tmp = gc; gc = gn; gn = tmp;
  }

  // ---- direct term: grad_x += 2*(x.wq_out)*wq_out + wl_out ; blend with x
  if (tid < ROWS) {
    float s = 0.0f;
    for (int d = 0; d < D_DIM; ++d) s += xs[tid * D_DIM + d] * wq_out[d];
    sbuf[tid] = s;
  }
  __syncthreads();
  for (int i = tid; i < ROWS * D_DIM; i += NTHREADS) {
    const int m = i / D_DIM;
    const int d = i - m * D_DIM;
    const float gx = gradx[i] + 2.0f * sbuf[m] * wq_out[d] + wl_out[d];
    out[row0 * D_DIM + i] = (1.0f - MU) * gx + MU * xs[i];
  }
}

extern "C" void kernel_launch(void* const* d_in, const int* in_sizes, int n_in,
                              void* d_out, int out_size, void* d_ws, size_t ws_size,
                              hipStream_t stream) {
  (void)in_sizes; (void)n_in; (void)out_size; (void)d_ws; (void)ws_size;
  const float* x      = (const float*)d_in[0];
  const float* Wq     = (const float*)d_in[1];
  const float* Wl     = (const float*)d_in[2];
  const float* bl     = (const float*)d_in[3];
  const float* Wz     = (const float*)d_in[4];
  const float* wz_out = (const float*)d_in[5];
  const float* wq_out = (const float*)d_in[6];
  const float* wl_out = (const float*)d_in[7];
  float* out = (float*)d_out;

  dim3 grid(B_TOT / ROWS);   // 512 workgroups of 16 rows
  dim3 block(NTHREADS);      // 8 waves (wave32)
  hipLaunchKernelGGL(icnn_grad_kernel, grid, block, SMEM_TOTAL, stream,
                     x, Wq, Wl, bl, Wz, wz_out, wq_out, wl_out, out);
}